// Hapke_83382495084809
// MI455X (gfx1250) — compile-verified
//
#include <hip/hip_runtime.h>
#include <math.h>

#define EPSF     1e-5f
#define PI_F     3.14159265358979323846f
#define INV_PI_F 0.3183098861837907f
#define TWO_OVER_PI_F 0.6366197723675814f

__device__ __forceinline__ float frcp(float x)  { return __builtin_amdgcn_rcpf(x);  }
__device__ __forceinline__ float frsq(float x)  { return __builtin_amdgcn_rsqf(x);  }
__device__ __forceinline__ float fsqrt(float x) { return __builtin_amdgcn_sqrtf(x); }
__device__ __forceinline__ float nan_to(float y, float rep) { return (y != y) ? rep : y; }
__device__ __forceinline__ float clamp1(float x) { return fminf(1.0f, fmaxf(-1.0f, x)); }
__device__ __forceinline__ float ntl(const float* p) { return __builtin_nontemporal_load(p); }

// _HF(x, w) = nan_to(1/(1 - w*x*(ro + (1-2*ro*x)/2 * log|(1+x)/x|)), 1)
__device__ __forceinline__ float HF(float x, float w) {
    float gamma = fsqrt(1.0f - w);
    float ro    = (1.0f - gamma) * frcp(1.0f + gamma);
    float lg    = __logf(fabsf((1.0f + x) * frcp(x)));
    float tmp1  = 1.0f - w * x * (ro + (1.0f - 2.0f * ro * x) * 0.5f * lg);
    return nan_to(frcp(tmp1), 1.0f);
}

// _PF(x, b, c): double Henyey-Greenstein; pow(d,1.5) = d*sqrt(d), d >= (1-b)^2 >= 0
__device__ __forceinline__ float PF(float x, float b, float c) {
    float b2 = b * b;
    float bx = b * x;
    float d1 = 1.0f - 2.0f * bx + b2;
    float d2 = 1.0f + 2.0f * bx + b2;
    float num = 1.0f - b2;
    float y = c          * num * frcp(d1 * fsqrt(d1) + 1e-6f)
            + (1.0f - c) * num * frcp(d2 * fsqrt(d2) + 1e-6f);
    return nan_to(y, 0.0f);
}

__global__ __launch_bounds__(256) void hapke_brdf_kernel(
    const float* __restrict__ pts2l,  // (N,1,3)
    const float* __restrict__ pts2c,  // (N,3)
    const float* __restrict__ normal, // (N,3)
    const float* __restrict__ w,      // (N,3)
    const float* __restrict__ b,      // (N,3)
    const float* __restrict__ c,      // (N,3)
    const float* __restrict__ theta,  // (N,)
    const float* __restrict__ h,      // (N,1)
    const float* __restrict__ B0,     // (N,1)
    float* __restrict__ out,          // (N,3)
    int n)
{
    int i = blockIdx.x * blockDim.x + threadIdx.x;
    if (i >= n) return;
    int i3 = 3 * i;

    // ---- geometry (per point) ----
    float lx = ntl(pts2l + i3),  ly = ntl(pts2l + i3 + 1),  lz = ntl(pts2l + i3 + 2);
    float vx = ntl(pts2c + i3),  vy = ntl(pts2c + i3 + 1),  vz = ntl(pts2c + i3 + 2);
    float nx = ntl(normal + i3), ny = ntl(normal + i3 + 1), nz = ntl(normal + i3 + 2);

    float ci = clamp1(nx * lx + ny * ly + nz * lz);   // cos(sza)
    float cv = clamp1(nx * vx + ny * vy + nz * vz);   // cos(vza)
    float cg = clamp1(lx * vx + ly * vy + lz * vz);   // cos(g)

    // sin(arccos(x)) == sqrt(1-x^2); exact identity, no acos/sin needed
    float si = fsqrt(fmaxf(0.0f, 1.0f - ci * ci));
    float sv = fsqrt(fmaxf(0.0f, 1.0f - cv * cv));

    float cphi = clamp1((cg - ci * cv) * frcp(si * sv + 1e-8f));
    float phi  = acosf(cphi);                          // only acos we still need
    float sphi = fsqrt(fmaxf(0.0f, 1.0f - cphi * cphi));

    // i <= e  <=>  ci >= cv  (acos strictly decreasing)
    bool ile = (ci >= cv);

    // ---- roughness terms (per point) ----
    float th   = ntl(theta + i);
    float tthe = __tanf(th + EPSF);          // tan(theta+EPS)
    float tth  = __tanf(th);                 // tan(theta) (no EPS, per _mu_eff)
    float cot_th = frcp(tthe);
    float chit = nan_to(frsq(1.0f + PI_F * tthe * tthe), 0.0f);   // _chi(theta)

    // cot(x+EPS) = cos(x+EPS)/sin(x+EPS) ~= (cx - sx*EPS)/(sx + cx*EPS)   [O(EPS^2)]
    float cot_i = (ci - si * EPSF) * frcp(si + ci * EPSF);
    float cot_e = (cv - sv * EPSF) * frcp(sv + cv * EPSF);

    float E1i = nan_to(__expf(-TWO_OVER_PI_F * cot_th * cot_i), 0.0f);
    float E1e = nan_to(__expf(-TWO_OVER_PI_F * cot_th * cot_e), 0.0f);
    float ct2 = cot_th * cot_th;
    float E2i = nan_to(__expf(-INV_PI_F * ct2 * cot_i * cot_i), 0.0f);
    float E2e = nan_to(__expf(-INV_PI_F * ct2 * cot_e * cot_e), 0.0f);

    float etai = nan_to(chit * (ci + si * tthe * (E2i * frcp(2.0f - E1i))), 0.0f);
    float etae = nan_to(chit * (cv + sv * tthe * (E2e * frcp(2.0f - E1e))), 0.0f);

    // ---- effective cosines (per point) ----
    float sp2  = 0.5f * (1.0f - cphi);            // sin^2(phi/2)
    float den1 = frcp(2.0f - E1e - phi * INV_PI_F * E1i);
    float den2 = frcp(2.0f - E1i - phi * INV_PI_F * E1e);

    float muy  = ile ? (E2e - sp2 * E2i) * den1
                     : (cphi * E2i + sp2 * E2e) * den2;
    float mue  = nan_to(chit * (cv + sv * tth * muy), cv);        // _mu_eff

    float mu0y = ile ? (cphi * E2e + sp2 * E2i) * den1
                     : (E2i - sp2 * E2e) * den2;
    float mu0  = nan_to(chit * (ci + si * tth * mu0y), ci);       // _mu0_eff

    // ---- shadowing S (per point) ----
    // tan((phi+EPS)/2) = sin(phi+EPS)/(1+cos(phi+EPS))
    //                 ~= (sphi + cphi*EPS)/(1 + cphi - sphi*EPS)   [O(EPS^2)]
    float tph2 = (sphi + cphi * EPSF) * frcp(1.0f + cphi - sphi * EPSF);
    float ff = nan_to(__expf(-2.0f * tph2), 0.0f);

    float ci_etai = ci * frcp(etai);
    float cv_etae = cv * frcp(etae);
    float temp = mue * frcp(etae) * ci_etai * chit;
    float sden = ile ? (1.0f - ff + ff * chit * ci_etai)
                     : (1.0f - ff + ff * chit * cv_etae);
    float Shad = nan_to(temp * frcp(sden), 0.0f);

    // ---- opposition surge B (per point) ----
    float hh  = ntl(h + i);
    float bb0 = ntl(B0 + i);
    // tan(g/2) = sqrt((1-cg)/(1+cg)), exact for g in [0,pi]; cg=-1 -> +inf -> B=1
    float tg2 = fsqrt(fmaxf(0.0f, (1.0f - cg) * frcp(1.0f + cg)));
    float Bv  = bb0 * frcp(1.0f + frcp(hh) * tg2) + 1.0f;

    float tmp1 = mu0 * frcp(mu0 + mue) * frcp(ci);
    float s4   = 0.25f * tmp1 * Shad;   // (1/HPK_SCL) * tmp1 * Shad

    // ---- per-channel ----
    #pragma unroll
    for (int k = 0; k < 3; ++k) {
        float wk = ntl(w + i3 + k);
        float bk = ntl(b + i3 + k);
        float ck = ntl(c + i3 + k);
        float P  = PF(cg, bk, ck);
        float Hi = HF(mu0, wk);
        float Hv = HF(mue, wk);
        float tmp2 = P * Bv + Hi * Hv - 1.0f;
        __builtin_nontemporal_store(wk * s4 * tmp2, out + i3 + k);
    }
}

extern "C" void kernel_launch(void* const* d_in, const int* in_sizes, int n_in,
                              void* d_out, int out_size, void* d_ws, size_t ws_size,
                              hipStream_t stream) {
    (void)n_in; (void)d_ws; (void)ws_size; (void)out_size;
    const float* pts2l  = (const float*)d_in[0];
    const float* pts2c  = (const float*)d_in[1];
    const float* normal = (const float*)d_in[2];
    const float* w      = (const float*)d_in[3];
    const float* b      = (const float*)d_in[4];
    const float* c      = (const float*)d_in[5];
    const float* theta  = (const float*)d_in[6];
    const float* h      = (const float*)d_in[7];
    const float* B0     = (const float*)d_in[8];
    float* out = (float*)d_out;

    int n = in_sizes[6];               // theta is (N,)
    int block = 256;                   // 8 wave32s
    int grid  = (n + block - 1) / block;
    hapke_brdf_kernel<<<grid, block, 0, stream>>>(pts2l, pts2c, normal, w, b, c,
                                                  theta, h, B0, out, n);
}